// KnowledgeSkipGram_6622839571290
// MI455X (gfx1250) — compile-verified
//
#include <hip/hip_runtime.h>
#include <hip/hip_bf16.h>

typedef __attribute__((ext_vector_type(2))) float v2f;
typedef __attribute__((ext_vector_type(4))) float v4f;
typedef __attribute__((ext_vector_type(8))) float v8f;

#define EMBED       512
#define NUM_SAMPLED 64

__device__ __forceinline__ float log_sigmoid(float x) {
    // stable: min(x,0) - log(1 + exp(-|x|)), exp argument <= 0 so 1+e in [1,2]
    return fminf(x, 0.f) - __logf(1.f + __expf(-fabsf(x)));
}

__global__ void zero_out_kernel(float* out) { out[0] = 0.f; }

__global__ __launch_bounds__(128)
void skipgram_loss_kernel(const int* __restrict__ input_labels,   // (B,2)
                          const int* __restrict__ center_noise,   // (B,64)
                          const float* __restrict__ word_w,       // (200000,512)
                          const float* __restrict__ doc_w,        // (50000,512)
                          float* __restrict__ out,
                          float inv_batch)
{
    __shared__ float sdoc[EMBED];
    __shared__ float s_pos;
    __shared__ float s_neg;

    const int b   = blockIdx.x;
    const int tid = threadIdx.x;
    const int center_id = input_labels[2 * b + 0];
    const int doc_id    = input_labels[2 * b + 1];

    if (tid == 0) { s_pos = 0.f; s_neg = 0.f; }

    // Stage doc row in LDS: 128 threads x float4 (coalesced 2 KB load)
    {
        v4f d = *(const v4f*)(doc_w + (size_t)doc_id * EMBED + tid * 4);
        *(v4f*)(sdoc + tid * 4) = d;
    }
    __syncthreads();

    // Positive logit: plain VALU dot (1 of 65 dots, not worth a matrix tile)
    {
        const float* wp = word_w + (size_t)center_id * EMBED + tid * 4;
        v4f wv = *(const v4f*)wp;
        v4f dv = *(const v4f*)(sdoc + tid * 4);
        float p = wv.x * dv.x + wv.y * dv.y + wv.z * dv.z + wv.w * dv.w;
        atomicAdd(&s_pos, p);   // ds_add_f32
    }

    // Negative logits: 4 waves x 16 rows, fp32 WMMA 16x16x4 with doc
    // vector broadcast across B's 16 columns.
    {
        const int wave = tid >> 5;
        const int lane = tid & 31;
        const int m    = lane & 15;               // A-matrix row owned by lane
        const int koff = (lane >= 16) ? 2 : 0;    // lanes 16-31 carry K=2,3

        const int nidx = center_noise[(size_t)b * NUM_SAMPLED + wave * 16 + m];
        const float* arow = word_w + (size_t)nidx * EMBED;

        v8f c = {0.f, 0.f, 0.f, 0.f, 0.f, 0.f, 0.f, 0.f};
        #pragma unroll 8
        for (int k0 = 0; k0 < EMBED; k0 += 4) {
            v2f a  = *(const v2f*)(arow + k0 + koff);   // global_load_b64 (gathered row)
            v2f bb = *(const v2f*)(sdoc + k0 + koff);   // ds_load_b64 (doc broadcast)
            c = __builtin_amdgcn_wmma_f32_16x16x4_f32(
                    /*neg_a=*/false, a, /*neg_b=*/false, bb,
                    /*c_mod=*/(short)0, c, /*reuse_a=*/false, /*reuse_b=*/false);
        }
        // C layout: lane 0 holds dots m=0..7 (col 0), lane 16 holds m=8..15.
        if (m == 0) {
            float t = 0.f;
            #pragma unroll
            for (int j = 0; j < 8; ++j) t += log_sigmoid(-c[j]);
            atomicAdd(&s_neg, t);
        }
    }
    __syncthreads();

    if (tid == 0) {
        float contrib = log_sigmoid(s_pos) + s_neg;
        atomicAdd(out, -contrib * inv_batch);   // global_atomic_add_f32
    }
}

extern "C" void kernel_launch(void* const* d_in, const int* in_sizes, int n_in,
                              void* d_out, int out_size, void* d_ws, size_t ws_size,
                              hipStream_t stream) {
    const int*   input_labels = (const int*)d_in[0];
    // d_in[1] = out_labels  (unused by the reference loss)
    const int*   center_noise = (const int*)d_in[2];
    // d_in[3] = num_sampled scalar (compile-time 64 here)
    const float* word_w       = (const float*)d_in[4];
    const float* doc_w        = (const float*)d_in[5];
    float*       out          = (float*)d_out;

    const int batch = in_sizes[1];          // out_labels has BATCH elements

    zero_out_kernel<<<1, 1, 0, stream>>>(out);
    skipgram_loss_kernel<<<batch, 128, 0, stream>>>(
        input_labels, center_noise, word_w, doc_w, out, 1.0f / (float)batch);
}